// Layer_35304631173426
// MI455X (gfx1250) — compile-verified
//
#include <hip/hip_runtime.h>
#include <math.h>

typedef float v2f __attribute__((ext_vector_type(2)));
typedef float v8f __attribute__((ext_vector_type(8)));

#define WMMA_F32(a, b, c) \
  __builtin_amdgcn_wmma_f32_16x16x4_f32(false, (a), false, (b), (short)0, (c), false, false)

constexpr int   N_NODES = 32768;
constexpr int   MUL     = 16;
constexpr int   TMUL    = 32;
constexpr float SCALE1  = 1.0f / 128.0f;              // (1/DENOM=1/32) * (1/sqrt(mul)=1/4)
constexpr float INV_IN  = 0.25f;                      // 1/sqrt(16)
constexpr float INV_MID = 0.17677669529663687f;       // 1/sqrt(32)
constexpr float C_GELU  = 1.5338061f;                 // 1/sqrt(E[gelu(z)^2]), z~N(0,1)

// SH constants
constexpr float S3    = 1.7320508075688772f;
constexpr float S5    = 2.23606797749979f;
constexpr float S15   = 3.872983346207417f;
constexpr float S7    = 2.6457513110645907f;
constexpr float S105  = 10.246950765959598f;
constexpr float S35_8 = 2.091650066335189f;
constexpr float S21_8 = 1.6201851746019653f;

// ---------------------------------------------------------------------------
// Edge kernel: one thread per (edge, mul-channel). 16 consecutive lanes share
// one edge -> their 16 atomics per component hit 16 consecutive floats.
// agg layout: agg[(c * N + node) * 16 + u]   (c = SH component 0..15)
// ---------------------------------------------------------------------------
__global__ __launch_bounds__(256) void edge_kernel(
    const float* __restrict__ x, const float* __restrict__ pos,
    const int* __restrict__ snd, const int* __restrict__ rcv,
    float* __restrict__ agg, int E) {
  int idx = blockIdx.x * 256 + threadIdx.x;
  int e = idx >> 4;
  int u = idx & 15;
  if (e >= E) return;

  int s = snd[e];
  int r = rcv[e];
  float dx = pos[r * 3 + 0] - pos[s * 3 + 0];
  float dy = pos[r * 3 + 1] - pos[s * 3 + 1];
  float dz = pos[r * 3 + 2] - pos[s * 3 + 2];
  float r2 = dx * dx + dy * dy + dz * dz;
  float inv = (r2 > 0.0f) ? rsqrtf(r2) : 0.0f;
  float X = dx * inv, Y = dy * inv, Z = dz * inv;

  float su = x[s * MUL + u];

  float xx = X * X, yy = Y * Y, zz = Z * Z;
  float sh[16];
  sh[0]  = 1.0f;                              // l=0
  sh[1]  = S3 * X;                            // l=1
  sh[2]  = S3 * Y;
  sh[3]  = S3 * Z;
  sh[4]  = S15 * X * Y;                       // l=2
  sh[5]  = S15 * Y * Z;
  sh[6]  = 0.5f * S5 * (3.0f * zz - 1.0f);
  sh[7]  = S15 * X * Z;
  sh[8]  = 0.5f * S15 * (xx - yy);
  sh[9]  = S35_8 * Y * (3.0f * xx - yy);      // l=3
  sh[10] = S105 * X * Y * Z;
  sh[11] = S21_8 * Y * (5.0f * zz - 1.0f);
  sh[12] = 0.5f * S7 * Z * (5.0f * zz - 3.0f);
  sh[13] = S21_8 * X * (5.0f * zz - 1.0f);
  sh[14] = 0.5f * S105 * Z * (xx - yy);
  sh[15] = S35_8 * X * (xx - 3.0f * yy);

  float* base = agg + r * MUL + u;
#pragma unroll
  for (int c = 0; c < 16; ++c) {
    unsafeAtomicAdd(base + c * (N_NODES * MUL), su * sh[c]);
  }
}

// ---------------------------------------------------------------------------
// Node kernel: one wave32 per (16-node tile, SH component c).
//   P = (agg_tile * 1/128) @ Wpre[l]          via 8x  v_wmma_f32_16x16x4_f32
//   l==0: P = C_GELU * gelu(P)
//   LDS transpose P: D-layout -> A-layout (wave-private, no barrier needed)
//   Q = P @ Wpost[l]                          via 16x v_wmma_f32_16x16x4_f32
//   c==0: Q += (x_tile * 1/4) @ Wsc           via 8x  v_wmma_f32_16x16x4_f32
// ---------------------------------------------------------------------------
__global__ __launch_bounds__(256) void node_kernel(
    const float* __restrict__ agg, const float* __restrict__ x,
    const float* __restrict__ Wpre,   // [4][16][32]
    const float* __restrict__ Wpost,  // [4][32][32]
    const float* __restrict__ Wsc,    // [16][32]
    float* __restrict__ out) {
  constexpr int RS = 36;  // padded LDS row stride (floats), keeps 8B alignment
  __shared__ float lds[8][16 * RS];

  int wave = threadIdx.x >> 5;
  int lane = threadIdx.x & 31;
  int tile = blockIdx.x * 8 + wave;
  int c  = tile & 15;          // SH component
  int n0 = (tile >> 4) * 16;   // node tile base

  int l, d_l, base_l, i_l;
  if (c == 0)      { l = 0; d_l = 1; base_l = 0;   i_l = 0;     }
  else if (c < 4)  { l = 1; d_l = 3; base_l = 32;  i_l = c - 1; }
  else if (c < 9)  { l = 2; d_l = 5; base_l = 128; i_l = c - 4; }
  else             { l = 3; d_l = 7; base_l = 288; i_l = c - 9; }

  int row = lane & 15;                 // M-row (stage A) / N-col (B,C,D)
  int kk  = (lane < 16) ? 0 : 2;       // half-wave K offset per WMMA layout

  // ---- stage 1: load A tile (rows = nodes, cols = u), pre-scaled ----
  const float* Arow = agg + ((size_t)c * N_NODES + n0 + row) * MUL;
  v2f a1[4];
#pragma unroll
  for (int k = 0; k < 4; ++k) {
    v2f t = *(const v2f*)(Arow + 4 * k + kk);
    a1[k].x = t.x * SCALE1;
    a1[k].y = t.y * SCALE1;
  }

  const float* Wp = Wpre + l * (MUL * TMUL);
  v8f p[2];
#pragma unroll
  for (int t = 0; t < 2; ++t) {
    v8f acc = {0.f, 0.f, 0.f, 0.f, 0.f, 0.f, 0.f, 0.f};
#pragma unroll
    for (int k = 0; k < 4; ++k) {
      int K = 4 * k + kk;
      int v = t * 16 + row;
      v2f b;
      b.x = Wp[K * TMUL + v];
      b.y = Wp[(K + 1) * TMUL + v];
      acc = WMMA_F32(a1[k], b, acc);
    }
    p[t] = acc;
  }

  // ---- exact gelu on l=0 scalars (wave-uniform branch: EXEC stays all-1) ----
  if (l == 0) {
#pragma unroll
    for (int t = 0; t < 2; ++t)
#pragma unroll
      for (int j = 0; j < 8; ++j) {
        float v = p[t][j];
        p[t][j] = C_GELU * 0.5f * v * (1.0f + erff(v * 0.7071067811865475f));
      }
  }

  // ---- D-layout -> A-layout transpose through wave-private LDS ----
  float* L = &lds[wave][0];
#pragma unroll
  for (int t = 0; t < 2; ++t)
#pragma unroll
    for (int rr = 0; rr < 8; ++rr) {
      int m   = rr + ((lane < 16) ? 0 : 8);  // node row of D element
      int col = t * 16 + row;                // v column
      L[m * RS + col] = p[t][rr];
    }

  v2f a2[8];
#pragma unroll
  for (int k2 = 0; k2 < 8; ++k2) {
    a2[k2] = *(const v2f*)(L + row * RS + 4 * k2 + kk);
  }

  // ---- stage 2: Q = P @ Wpost[l] ----
  const float* Wq = Wpost + l * (TMUL * TMUL);
  v8f q[2];
#pragma unroll
  for (int t = 0; t < 2; ++t) {
    v8f acc = {0.f, 0.f, 0.f, 0.f, 0.f, 0.f, 0.f, 0.f};
#pragma unroll
    for (int k2 = 0; k2 < 8; ++k2) {
      int K = 4 * k2 + kk;
      int w = t * 16 + row;
      v2f b;
      b.x = Wq[K * TMUL + w];
      b.y = Wq[(K + 1) * TMUL + w];
      acc = WMMA_F32(a2[k2], b, acc);
    }
    q[t] = acc;
  }

  // ---- shortcut (only the l=0 / c==0 waves): (x * 1/4) @ Wsc ----
  v8f sc[2] = {{0.f, 0.f, 0.f, 0.f, 0.f, 0.f, 0.f, 0.f},
               {0.f, 0.f, 0.f, 0.f, 0.f, 0.f, 0.f, 0.f}};
  if (c == 0) {
    const float* Xrow = x + (size_t)(n0 + row) * MUL;
    v2f ax[4];
#pragma unroll
    for (int k = 0; k < 4; ++k) {
      v2f t = *(const v2f*)(Xrow + 4 * k + kk);
      ax[k].x = t.x * INV_IN;
      ax[k].y = t.y * INV_IN;
    }
#pragma unroll
    for (int t = 0; t < 2; ++t) {
      v8f acc = {0.f, 0.f, 0.f, 0.f, 0.f, 0.f, 0.f, 0.f};
#pragma unroll
      for (int k = 0; k < 4; ++k) {
        int K = 4 * k + kk;
        int w = t * 16 + row;
        v2f b;
        b.x = Wsc[K * TMUL + w];
        b.y = Wsc[(K + 1) * TMUL + w];
        acc = WMMA_F32(ax[k], b, acc);
      }
      sc[t] = acc;
    }
  }

  // ---- store: out[node, base_l + w*d_l + i_l] ----
#pragma unroll
  for (int t = 0; t < 2; ++t)
#pragma unroll
    for (int rr = 0; rr < 8; ++rr) {
      int node = n0 + rr + ((lane < 16) ? 0 : 8);
      int w    = t * 16 + row;
      out[(size_t)node * 512 + base_l + w * d_l + i_l] =
          q[t][rr] * INV_MID + sc[t][rr];
    }
}

// ---------------------------------------------------------------------------
extern "C" void kernel_launch(void* const* d_in, const int* in_sizes, int n_in,
                              void* d_out, int out_size, void* d_ws, size_t ws_size,
                              hipStream_t stream) {
  const float* x     = (const float*)d_in[0];
  const float* pos   = (const float*)d_in[1];
  const float* Wpre  = (const float*)d_in[2];
  const float* Wpost = (const float*)d_in[3];
  const float* Wsc   = (const float*)d_in[4];
  const int*   snd   = (const int*)d_in[5];
  const int*   rcv   = (const int*)d_in[6];
  int E = in_sizes[5];

  float* agg = (float*)d_ws;  // [16 comps][N][16 mul] = 32 MB, L2-resident
  size_t aggBytes = (size_t)16 * N_NODES * MUL * sizeof(float);
  hipMemsetAsync(agg, 0, aggBytes, stream);

  int ethreads = E * 16;
  edge_kernel<<<(ethreads + 255) / 256, 256, 0, stream>>>(x, pos, snd, rcv, agg, E);

  int tiles = (N_NODES / 16) * 16;  // 32768 waves, 8 per block
  node_kernel<<<tiles / 8, 256, 0, stream>>>(agg, x, Wpre, Wpost, Wsc, (float*)d_out);
}